// DCPLoss_13297218748829
// MI455X (gfx1250) — compile-verified
//
#include <hip/hip_runtime.h>
#include <cstdint>
#include <cstddef>

#define IMG 640
#define HW (640 * 640)

typedef __attribute__((ext_vector_type(2))) float v2f;
typedef __attribute__((ext_vector_type(8))) float v8f;

// ---------------- CDNA5-specific helpers ----------------

__device__ __forceinline__ v8f wmma_f32_16x16x4(v2f a, v2f b, v8f c) {
#if defined(__gfx1250__) && __has_builtin(__builtin_amdgcn_wmma_f32_16x16x4_f32)
  // 8 args: (neg_a, A, neg_b, B, c_mod, C, reuse_a, reuse_b)
  return __builtin_amdgcn_wmma_f32_16x16x4_f32(false, a, false, b, (short)0, c,
                                               false, false);
#else
  (void)a; (void)b;
  return c;  // host-pass / non-gfx1250 stub (never executed on device)
#endif
}

__device__ __forceinline__ void async_ld_f32(const float* g, float* lds) {
#if defined(__gfx1250__) && __has_builtin(__builtin_amdgcn_global_load_async_to_lds_b32)
  __builtin_amdgcn_global_load_async_to_lds_b32(
      (__attribute__((address_space(1))) int*)const_cast<float*>(g),
      (__attribute__((address_space(3))) int*)lds, 0, 0);
#else
  *lds = *g;
#endif
}

__device__ __forceinline__ void async_wait0() {
#if defined(__gfx1250__) && __has_builtin(__builtin_amdgcn_s_wait_asynccnt)
  __builtin_amdgcn_s_wait_asynccnt(0);
#elif defined(__gfx1250__)
  asm volatile("s_wait_asynccnt 0" ::: "memory");
#endif
}

__device__ __forceinline__ float band7(int d) {
  return (d >= 0 && d <= 6) ? 1.0f : 0.0f;
}

// ---------------- K1: dark channel (quantize + ch-min + 15x15 erode) -------
// grid (10,10,24): 64x64 output tile per block; map = tensor*8 + frame.
__global__ __launch_bounds__(256) void k1_dark(const float* __restrict__ gt,
                                               const float* __restrict__ op,
                                               const float* __restrict__ em,
                                               float* __restrict__ dark) {
  __shared__ float sm[78 * 80];  // channel-min tile + 7px halo, stride 80
  __shared__ float sh[78 * 64];  // after horizontal min-15
  const int map = blockIdx.z;
  const float* src = (map < 8) ? gt : (map < 16) ? op : em;
  const float* base = src + (size_t)(map & 7) * 3 * HW;
  const int ox = blockIdx.x * 64 - 7;
  const int oy = blockIdx.y * 64 - 7;

  for (int i = threadIdx.x; i < 78 * 78; i += 256) {
    const int r = i / 78, c = i - r * 78;
    const int gy = oy + r, gx = ox + c;
    float q;
    if ((unsigned)gy < 640u && (unsigned)gx < 640u) {
      const int idx = gy * IMG + gx;
      float m = fminf(base[idx], fminf(base[HW + idx], base[2 * HW + idx]));
      q = floorf(fminf(fmaxf(m, 0.0f), 1.0f) * 255.0f);  // == byte quantize
    } else {
      q = __builtin_inff();  // cv2.erode border value
    }
    sm[r * 80 + c] = q;
  }
  __syncthreads();

  for (int i = threadIdx.x; i < 78 * 64; i += 256) {
    const int r = i >> 6, x = i & 63;
    float m = sm[r * 80 + x];
#pragma unroll
    for (int d = 1; d < 15; ++d) m = fminf(m, sm[r * 80 + x + d]);
    sh[i] = m;
  }
  __syncthreads();

  for (int i = threadIdx.x; i < 64 * 64; i += 256) {
    const int y = i >> 6, x = i & 63;
    float m = sh[y * 64 + x];
#pragma unroll
    for (int d = 1; d < 15; ++d) m = fminf(m, sh[(y + d) * 64 + x]);
    const int gy = blockIdx.y * 64 + y, gx = blockIdx.x * 64 + x;
    dark[(size_t)map * HW + gy * IMG + gx] = m * (1.0f / 255.0f);
  }
}

// ---------------- K2: temporal residuals + global var partials -------------
__global__ __launch_bounds__(256) void k2_residual(const float* __restrict__ dark,
                                                   float* __restrict__ rsr,
                                                   float* __restrict__ rema,
                                                   float* __restrict__ part1) {
  const int i = blockIdx.x * 256 + threadIdx.x;  // < HW
  const float* dg = dark;
  const float* dq = dark + (size_t)8 * HW;
  const float* de = dark + (size_t)16 * HW;
  float a = 0.0f, b = 0.0f;
#pragma unroll
  for (int t = 0; t < 8; ++t) {
    const float g = dg[t * HW + i];
    a += fabsf(g - dq[t * HW + i]);
    b += fabsf(g - de[t * HW + i]);
  }
  rsr[i] = a;
  rema[i] = b;

  __shared__ float s1[256], s2[256];
  s1[threadIdx.x] = a;
  s2[threadIdx.x] = a * a;
  __syncthreads();
  for (int off = 128; off > 0; off >>= 1) {
    if ((int)threadIdx.x < off) {
      s1[threadIdx.x] += s1[threadIdx.x + off];
      s2[threadIdx.x] += s2[threadIdx.x + off];
    }
    __syncthreads();
  }
  if (threadIdx.x == 0) {
    part1[2 * blockIdx.x] = s1[0];
    part1[2 * blockIdx.x + 1] = s2[0];
  }
}

// ---------------- K3: patch_w = Var(rsr, ddof=1)^0.2 -----------------------
__global__ __launch_bounds__(256) void k3_patchw(const float* __restrict__ part1,
                                                 float* __restrict__ patchw) {
  __shared__ float s1[256], s2[256];
  float a = 0.0f, b = 0.0f;
  for (int i = threadIdx.x; i < 1600; i += 256) {
    a += part1[2 * i];
    b += part1[2 * i + 1];
  }
  s1[threadIdx.x] = a;
  s2[threadIdx.x] = b;
  __syncthreads();
  for (int off = 128; off > 0; off >>= 1) {
    if ((int)threadIdx.x < off) {
      s1[threadIdx.x] += s1[threadIdx.x + off];
      s2[threadIdx.x] += s2[threadIdx.x + off];
    }
    __syncthreads();
  }
  if (threadIdx.x == 0) {
    const float N = 409600.0f;
    const float var = (s2[0] - s1[0] * s1[0] / N) / (N - 1.0f);
    patchw[0] = powf(var, 0.2f);
  }
}

// ---------------- K4: 7x7 box sums via WMMA, mask, L1 accumulation ---------
// One wave (32 threads) per 16x16 pixel tile; grid = 1600 blocks of 32.
// S = Av(16x22) * P(22x22) * Ah(22x16) with band-of-ones matrices, done as
// K=4 chunked V_WMMA_F32_16X16X4_F32 ops. Padding is PHYSICAL (P is 32x24
// with zeroed pad cells) so every LDS A-operand fetch is an unconditional
// ds_load_b64 -- no divergent saveexec around the WMMA feed.
//
// M1 layout: row-pair interleaved, M1[(row>>1)*32 + col*2 + (row&1)], so the
// stage-2 B operand (rows kk, kk+1 at column n) is one ds_load_b64.
__device__ v8f box7_sums(const float* P, float* M1, int lane, bool squared) {
  const int half = lane >> 4;   // K half / M half selector per WMMA layout
  const int mn = lane & 15;     // row (A) / column (B,D) index
  const v8f zero = {0.f, 0.f, 0.f, 0.f, 0.f, 0.f, 0.f, 0.f};

  // Stage 1: M1(32x16; rows 22..31 become true zeros) = P(32x24) x Ah(24x16)
  v8f dt = zero, db = zero;
#pragma unroll
  for (int k0 = 0; k0 < 24; k0 += 4) {
    const int kk = k0 + 2 * half;
    float ax = P[mn * 24 + kk];
    float ay = P[mn * 24 + kk + 1];
    float cx = P[(16 + mn) * 24 + kk];
    float cy = P[(16 + mn) * 24 + kk + 1];
    if (squared) { ax *= ax; ay *= ay; cx *= cx; cy *= cy; }
    v2f a_top, a_bot, b;
    a_top.x = ax; a_top.y = ay;
    a_bot.x = cx; a_bot.y = cy;
    b.x = band7(kk - mn);        // Ah[j][n] = 1 iff 0 <= j-n <= 6
    b.y = band7(kk + 1 - mn);
    dt = wmma_f32_16x16x4(a_top, b, dt);
    db = wmma_f32_16x16x4(a_bot, b, db);
  }
  // Spill D layout (VGPR r -> row r+8*half) to pair-interleaved M1
#pragma unroll
  for (int r = 0; r < 8; ++r) {
    const int m = r + 8 * half;            // dt row m, db row 16+m
    M1[(m >> 1) * 32 + mn * 2 + (m & 1)] = dt[r];
    M1[(8 + (m >> 1)) * 32 + mn * 2 + (m & 1)] = db[r];
  }
  // Stage 2: S = Av(16x24) x M1(24x16)
  v8f s = zero;
#pragma unroll
  for (int k0 = 0; k0 < 24; k0 += 4) {
    const int kk = k0 + 2 * half;
    v2f a, b;
    a.x = band7(kk - mn);        // Av[m][j] = 1 iff 0 <= j-m <= 6
    a.y = band7(kk + 1 - mn);
    b.x = M1[(kk >> 1) * 32 + mn * 2];      // row kk,  col mn
    b.y = M1[(kk >> 1) * 32 + mn * 2 + 1];  // row kk+1, col mn
    s = wmma_f32_16x16x4(a, b, s);
  }
  return s;
}

__global__ __launch_bounds__(32) void k4_pixel(const float* __restrict__ rsr,
                                               const float* __restrict__ rema,
                                               const float* __restrict__ gt,
                                               const float* __restrict__ op,
                                               float* __restrict__ part2) {
  __shared__ float P[32 * 24];   // reflect-padded tile + physical zero pad
  __shared__ float M1[16 * 32];  // stage-1 intermediate, pair-interleaved
  const int lane = threadIdx.x;  // single wave32 per block
  const int tile = blockIdx.x;   // 40x40 tiles of 16x16
  const int ty = tile / 40, tx = tile - ty * 40;
  const int gy0 = ty * 16, gx0 = tx * 16;

  // Zero only the pad cells (disjoint from the async-loaded 22x22 region,
  // so no DS-store vs async-LDS-write ordering hazard):
  for (int i = lane; i < 240; i += 32) P[22 * 24 + i] = 0.0f;  // rows 22..31
  {
    const int r = lane >> 1, c = 22 + (lane & 1);               // cols 22,23
    P[r * 24 + c] = 0.0f;                                       // rows 0..15
    if (r + 16 < 22) P[(r + 16) * 24 + c] = 0.0f;               // rows 16..21
  }

  // Gather the 22x22 reflect-padded residual tile via async-to-LDS.
  for (int i = lane; i < 22 * 22; i += 32) {
    const int r = i / 22, c = i - r * 22;
    int gy = gy0 - 3 + r;
    gy = (gy < 0) ? -gy : ((gy > 639) ? 1278 - gy : gy);
    int gx = gx0 - 3 + c;
    gx = (gx < 0) ? -gx : ((gx > 639) ? 1278 - gx : gx);
    async_ld_f32(rsr + gy * IMG + gx, P + r * 24 + c);
  }
  async_wait0();
  __syncthreads();

  const v8f s1 = box7_sums(P, M1, lane, false);
  const v8f s2 = box7_sums(P, M1, lane, true);

  const int half = lane >> 4, mn = lane & 15;
  float acc = 0.0f;
#pragma unroll
  for (int r = 0; r < 8; ++r) {
    const int m = r + 8 * half;
    const int gy = gy0 + m, gx = gx0 + mn;
    const float sv = s1[r];
    const float var = (s2[r] - sv * sv * (1.0f / 49.0f)) * (1.0f / 48.0f);
    const float rc = P[(m + 3) * 24 + (mn + 3)];  // center residual_sr
    const float re = rema[gy * IMG + gx];
    const float w = (rc < re) ? 0.0f : fabsf(var);  // |w| since |w*o - w*g|
    if (w != 0.0f) {
      const float* g = gt + gy * IMG + gx;
      const float* o = op + gy * IMG + gx;
      float S = 0.0f;
#pragma unroll
      for (int q = 0; q < 24; ++q) S += fabsf(o[q * HW] - g[q * HW]);
      acc += w * S;
    }
  }
  for (int off = 16; off > 0; off >>= 1) acc += __shfl_down(acc, off, 32);
  if (lane == 0) part2[blockIdx.x] = acc;
}

// ---------------- K5: final reduce + scale ---------------------------------
__global__ __launch_bounds__(256) void k5_final(const float* __restrict__ part2,
                                                const float* __restrict__ patchw,
                                                float* __restrict__ out) {
  __shared__ float s1[256];
  float a = 0.0f;
  for (int i = threadIdx.x; i < 1600; i += 256) a += part2[i];
  s1[threadIdx.x] = a;
  __syncthreads();
  for (int off = 128; off > 0; off >>= 1) {
    if ((int)threadIdx.x < off) s1[threadIdx.x] += s1[threadIdx.x + off];
    __syncthreads();
  }
  if (threadIdx.x == 0) {
    // mean over 1*8*3*640*640 elements; LDL_WEIGHT = 1
    out[0] = patchw[0] * s1[0] * (1.0f / 9830400.0f);
  }
}

// ---------------- launcher -------------------------------------------------
extern "C" void kernel_launch(void* const* d_in, const int* in_sizes, int n_in,
                              void* d_out, int out_size, void* d_ws,
                              size_t ws_size, hipStream_t stream) {
  (void)in_sizes; (void)n_in; (void)out_size; (void)ws_size;
  const float* gt = (const float*)d_in[0];
  const float* op = (const float*)d_in[1];
  const float* em = (const float*)d_in[2];
  float* out = (float*)d_out;

  float* ws = (float*)d_ws;
  float* dark = ws;                         // 24 * HW
  float* rsr = dark + (size_t)24 * HW;      // HW
  float* rema = rsr + HW;                   // HW
  float* part1 = rema + HW;                 // 2 * 1600
  float* patchw = part1 + 3200;             // 1
  float* part2 = patchw + 1;                // 1600

  dim3 g1(10, 10, 24);
  k1_dark<<<g1, 256, 0, stream>>>(gt, op, em, dark);
  k2_residual<<<1600, 256, 0, stream>>>(dark, rsr, rema, part1);
  k3_patchw<<<1, 256, 0, stream>>>(part1, patchw);
  k4_pixel<<<1600, 32, 0, stream>>>(rsr, rema, gt, op, part2);
  k5_final<<<1, 256, 0, stream>>>(part2, patchw, out);
}